// Predictor_20675972563129
// MI455X (gfx1250) — compile-verified
//
#include <hip/hip_runtime.h>
#include <hip/hip_bf16.h>

typedef __attribute__((ext_vector_type(2))) float v2f;
typedef __attribute__((ext_vector_type(8))) float v8f;

#define BQ 512
#define SQ 1024
#define DQ 24
#define HQ 64
#define NHQ 192  // 3*H

// V_WMMA_F32_16X16X4_F32: D(16x16 f32, v8f) = A(16x4 f32, v2f) * B(4x16 f32, v2f) + C
// 8-arg form: (neg_a, A, neg_b, B, c_mod, C, reuse_a, reuse_b)
static __device__ __forceinline__ v8f wmma4(v2f a, v2f b, v8f c) {
  return __builtin_amdgcn_wmma_f32_16x16x4_f32(false, a, false, b, (short)0, c,
                                               false, false);
}

__global__ __launch_bounds__(128, 1)
void gru_fused_kernel(const float* __restrict__ x,   // (B,S,D)
                      const int*   __restrict__ t,   // (B,)
                      const float* __restrict__ W,   // (D,3H)
                      const float* __restrict__ U,   // (H,3H)
                      const float* __restrict__ bias,// (2,3H)
                      const float* __restrict__ Wd,  // (H,1)
                      const float* __restrict__ bd,  // (1,)
                      float*       __restrict__ out) // (B,S,1)
{
  __shared__ float sh[16 * 68];   // h tile, padded stride 68 (conflict-free b64 A-loads)
  __shared__ float sx[16 * 26];   // x_s tile, padded stride 26
  __shared__ float sred[64];      // per-wave partial Wd dots

  const int tid  = threadIdx.x;
  const int wv   = tid >> 5;      // wave 0..3
  const int ln   = tid & 31;
  const int half = ln >> 4;       // 0: lanes 0-15, 1: lanes 16-31
  const int l16  = ln & 15;
  const int b0   = blockIdx.x * 16;

  // this wave's column within each gate block (z, r, c) and its h-column
  const int nz   = 16 * wv + l16;
  const int nr   = 64 + 16 * wv + l16;
  const int nc   = 128 + 16 * wv + l16;
  const int jcol = 16 * wv + l16;

  // ---- load U, W into registers in WMMA B-operand layout ----
  // B(4x16): VGPR g, lane l -> B[K = g + 2*(l/16)][N = l%16]
  v2f Uz[16], Ur[16], Uc[16];
#pragma unroll
  for (int kc = 0; kc < 16; ++kc) {
    const int k0 = kc * 4 + 2 * half;
    v2f a, b, c;
    a.x = U[(k0 + 0) * NHQ + nz]; a.y = U[(k0 + 1) * NHQ + nz];
    b.x = U[(k0 + 0) * NHQ + nr]; b.y = U[(k0 + 1) * NHQ + nr];
    c.x = U[(k0 + 0) * NHQ + nc]; c.y = U[(k0 + 1) * NHQ + nc];
    Uz[kc] = a; Ur[kc] = b; Uc[kc] = c;
  }
  v2f Wz[6], Wr[6], Wc[6];
#pragma unroll
  for (int kc = 0; kc < 6; ++kc) {
    const int k0 = kc * 4 + 2 * half;
    v2f a, b, c;
    a.x = W[(k0 + 0) * NHQ + nz]; a.y = W[(k0 + 1) * NHQ + nz];
    b.x = W[(k0 + 0) * NHQ + nr]; b.y = W[(k0 + 1) * NHQ + nr];
    c.x = W[(k0 + 0) * NHQ + nc]; c.y = W[(k0 + 1) * NHQ + nc];
    Wz[kc] = a; Wr[kc] = b; Wc[kc] = c;
  }

  // gate biases (broadcast along M, so one scalar per lane per gate)
  const float bz  = bias[nz] + bias[NHQ + nz];  // b0_z + b1_z
  const float br  = bias[nr] + bias[NHQ + nr];  // b0_r + b1_r
  const float bxh = bias[nc];                   // b0_c (goes with x@W)
  const float bhh = bias[NHQ + nc];             // b1_c (goes with h@U, scaled by r)
  const float wd  = Wd[jcol];
  const float bdv = bd[0];

  // sequence lengths for the 8 batch rows each lane touches (one per vgpr)
  int tv[8];
#pragma unroll
  for (int v = 0; v < 8; ++v) tv[v] = t[b0 + v + 8 * half];

  // h register copy in C/D layout: hreg[v] = h[M = v + 8*half][N = jcol]
  v8f hreg;
#pragma unroll
  for (int v = 0; v < 8; ++v) hreg[v] = 0.0f;

  for (int i = tid; i < 16 * 68; i += 128) sh[i] = 0.0f;
  __syncthreads();

  for (int s = 0; s < SQ; ++s) {
    // ---- stage x[b0:b0+16, s, :] into LDS (coalesced-ish, 384 = 3*128) ----
#pragma unroll
    for (int r = 0; r < 3; ++r) {
      const int idx = tid + r * 128;
      const int row = idx / 24;
      const int col = idx - row * 24;
      sx[row * 26 + col] = x[((size_t)(b0 + row) * SQ + s) * DQ + col];
    }
    __syncthreads();  // barrier A: x staged; prev-step h stores visible

    v8f accz, accr, accxh, acchh;
#pragma unroll
    for (int v = 0; v < 8; ++v) {
      accz[v] = bz; accr[v] = br; accxh[v] = bxh; acchh[v] = bhh;
    }

    // ---- h @ U  (K = 64 -> 16 chunks of 4) ----
    // A(16x4): VGPR g, lane l -> A[M = l%16][K = g + 2*(l/16)] -> one b64 load
#pragma unroll
    for (int kc = 0; kc < 16; ++kc) {
      const int k0 = kc * 4 + 2 * half;
      v2f a = *reinterpret_cast<const v2f*>(&sh[l16 * 68 + k0]);
      accz  = wmma4(a, Uz[kc], accz);
      accr  = wmma4(a, Ur[kc], accr);
      acchh = wmma4(a, Uc[kc], acchh);
    }
    // ---- x_s @ W (K = 24 -> 6 chunks of 4) ----
#pragma unroll
    for (int kc = 0; kc < 6; ++kc) {
      const int k0 = kc * 4 + 2 * half;
      v2f a = *reinterpret_cast<const v2f*>(&sx[l16 * 26 + k0]);
      accz  = wmma4(a, Wz[kc], accz);
      accr  = wmma4(a, Wr[kc], accr);
      accxh = wmma4(a, Wc[kc], accxh);
    }

    // ---- gates + masked update + fused Wd dot (partial per wave) ----
    float red[8];
#pragma unroll
    for (int v = 0; v < 8; ++v) {
      const float z  = 1.0f / (1.0f + __expf(-accz[v]));
      const float r  = 1.0f / (1.0f + __expf(-accr[v]));
      const float c  = tanhf(accxh[v] + r * acchh[v]);
      float hn = z * hreg[v] + (1.0f - z) * c;
      hn = (s < tv[v]) ? hn : hreg[v];   // masked step keeps carried state
      hreg[v] = hn;
      float p = hn * wd;                 // reduce over the 16 N-lanes
      p += __shfl_xor(p, 1, 32);
      p += __shfl_xor(p, 2, 32);
      p += __shfl_xor(p, 4, 32);
      p += __shfl_xor(p, 8, 32);
      red[v] = p;
    }
    if (l16 == 0) {
#pragma unroll
      for (int v = 0; v < 8; ++v) sred[wv * 16 + v + 8 * half] = red[v];
    }
    __syncthreads();  // barrier B: all sh A-reads done; sred complete

    // write updated h back to LDS for next step's A operand
#pragma unroll
    for (int v = 0; v < 8; ++v) sh[(v + 8 * half) * 68 + jcol] = hreg[v];

    // head: out[b, s] = sigmoid(h . Wd + bd)
    if (tid < 16) {
      const float acc =
          bdv + sred[tid] + sred[16 + tid] + sred[32 + tid] + sred[48 + tid];
      out[(size_t)(b0 + tid) * SQ + s] = 1.0f / (1.0f + __expf(-acc));
    }
  }
}

extern "C" void kernel_launch(void* const* d_in, const int* in_sizes, int n_in,
                              void* d_out, int out_size, void* d_ws, size_t ws_size,
                              hipStream_t stream) {
  const float* x  = (const float*)d_in[0];
  const int*   t  = (const int*)d_in[1];
  const float* W  = (const float*)d_in[2];
  const float* U  = (const float*)d_in[3];
  const float* b  = (const float*)d_in[4];
  const float* Wd = (const float*)d_in[5];
  const float* bd = (const float*)d_in[6];
  float* out = (float*)d_out;
  (void)in_sizes; (void)n_in; (void)out_size; (void)d_ws; (void)ws_size;

  dim3 grid(BQ / 16);
  dim3 block(128);
  hipLaunchKernelGGL(gru_fused_kernel, grid, block, 0, stream,
                     x, t, W, U, b, Wd, bd, out);
}